// ScaleDotProductAttention_8392366096729
// MI455X (gfx1250) — compile-verified
//
#include <hip/hip_runtime.h>
#include <math.h>

typedef float v2f __attribute__((ext_vector_type(2)));
typedef float v8f __attribute__((ext_vector_type(8)));

#define BB 2
#define HH 16
#define LL 2048
#define DD 64
#define TQ 16
#define SROW 2050   // padded S row stride in floats (2050 % 64 == 2 -> conflict-free A reads)
#define QROW 66     // padded Q row stride in floats

__launch_bounds__(256, 1)
__global__ void attn_f32_wmma_kernel(const float* __restrict__ q,
                                     const float* __restrict__ k,
                                     const float* __restrict__ v,
                                     const int*   __restrict__ mask,
                                     float* __restrict__ out,    // [B,H,L,D]
                                     float* __restrict__ score)  // [B,H,L,L]
{
    __shared__ float S[TQ * SROW];          // ~131 KB: S tile then P tile
    __shared__ float Qs[TQ * QROW];         // ~4.2 KB
    __shared__ float Os[8 * 2 * TQ * 16];   // 16 KB: per-wave 2 partial O tiles

    const int tid  = threadIdx.x;
    const int wave = tid >> 5;
    const int lane = tid & 31;
    const int kh   = lane >> 4;   // half-wave: selects K pair (A/B) / M+8 (C,D)
    const int l16  = lane & 15;   // A: M row ; B: N col ; C/D: N col

    const int bh    = blockIdx.x / (LL / TQ);   // 0..B*H-1
    const int qtile = blockIdx.x % (LL / TQ);
    const int b     = bh / HH;
    const int q0    = qtile * TQ;

    const float* qbase = q + ((size_t)bh * LL + q0) * DD;
    const float* kbase = k + (size_t)bh * LL * DD;
    const float* vbase = v + (size_t)bh * LL * DD;
    const int*   mbase = mask + ((size_t)b * LL + q0) * LL;   // mask[B,1,L,L]
    float*       obase = out + ((size_t)bh * LL + q0) * DD;
    float*       sbase = score + ((size_t)bh * LL + q0) * LL;

    // ---- stage Q tile (16x64) into LDS with padded stride ----
    for (int i = tid; i < TQ * DD; i += 256) {
        int m = i / DD, d = i % DD;
        Qs[m * QROW + d] = qbase[m * DD + d];
    }
    __syncthreads();

    // ---- S = Q @ K^T : wave w handles k-tiles w, w+8, ... (128 total) ----
    // (compiler hoists the 32 Q A-operand VGPRs out of the kt loop)
    for (int kt = wave; kt < LL / 16; kt += 8) {
        v8f acc = {};
        const float* krow = kbase + (size_t)(kt * 16 + l16) * DD; // K row n (B operand)
        const float* qrow = Qs + l16 * QROW;                      // Q row m (A operand)
        #pragma unroll
        for (int dd = 0; dd < DD / 4; ++dd) {
            const int d0 = dd * 4 + kh * 2;
            v2f a;  a.x  = qrow[d0]; a.y  = qrow[d0 + 1];
            v2f bm; bm.x = krow[d0]; bm.y = krow[d0 + 1];
            acc = __builtin_amdgcn_wmma_f32_16x16x4_f32(false, a, false, bm,
                                                        (short)0, acc, false, false);
        }
        #pragma unroll
        for (int r = 0; r < 8; ++r) {
            const int m = r + 8 * kh;                 // C/D layout: M = r + 8*half
            S[m * SROW + kt * 16 + l16] = acc[r];
        }
    }
    __syncthreads();

    // ---- softmax over rows: scale, mask, max, exp, sum, normalize ----
    {
        const int row = tid >> 4;       // 16 rows, 16 threads each
        const int cl  = tid & 15;
        float* srow = S + row * SROW;
        const int* mrow = mbase + (size_t)row * LL;

        // pass 1: apply scale+mask in place, find row max (mask: NT load, read once)
        float mx = -INFINITY;
        for (int jj = 0; jj < LL / 16; ++jj) {
            const int c = cl + 16 * jj;
            const int mv = __builtin_nontemporal_load(&mrow[c]);
            const float s = (mv == 0) ? -INFINITY : srow[c] * 0.125f;
            srow[c] = s;
            mx = fmaxf(mx, s);
        }
        #pragma unroll
        for (int off = 8; off >= 1; off >>= 1)
            mx = fmaxf(mx, __shfl_xor(mx, off, 32));
        const bool dead = (mx == -INFINITY);          // fully masked row -> 0s

        // pass 2: exp in place, row sum
        float sum = 0.0f;
        for (int jj = 0; jj < LL / 16; ++jj) {
            const int c = cl + 16 * jj;
            const float e = dead ? 0.0f : __expf(srow[c] - mx);
            srow[c] = e;
            sum += e;
        }
        #pragma unroll
        for (int off = 8; off >= 1; off >>= 1)
            sum += __shfl_xor(sum, off, 32);
        const float inv = dead ? 0.0f : (1.0f / sum);

        // pass 3: normalize; keep P in LDS, stream score to global with NT stores
        // (score is write-once: keep it out of L2 so K/V stay resident)
        float* srow_out = sbase + (size_t)row * LL;
        for (int jj = 0; jj < LL / 16; ++jj) {
            const int c = cl + 16 * jj;
            const float p = srow[c] * inv;
            srow[c] = p;
            __builtin_nontemporal_store(p, &srow_out[c]);
        }
    }
    __syncthreads();

    // ---- O = P @ V : wave -> 2 n-tiles ((w&1)*2, (w&1)*2+1) x K-quarter (w>>1) ----
    // Each A-operand LDS load feeds 2 WMMAs (halves LDS read traffic vs 1 tile/wave).
    {
        const int tgrp  = wave & 1;                   // n-tile group: tiles {0,1} or {2,3}
        const int kq    = wave >> 1;                  // K quarter: 512 rows
        const int n0    = tgrp * 32 + l16;            // V column for local tile 0
        const int n1    = n0 + 16;                    // V column for local tile 1
        v8f acc0 = {}, acc1 = {};
        const float* prow = S + l16 * SROW;           // A operand: P row m = l16
        #pragma unroll 4
        for (int kb = kq * (LL / 4); kb < (kq + 1) * (LL / 4); kb += 4) {
            const int k0 = kb + kh * 2;
            v2f a;  a.x = prow[k0]; a.y = prow[k0 + 1];
            const float* vr0 = vbase + (size_t)k0 * DD;
            const float* vr1 = vr0 + DD;
            v2f b0; b0.x = vr0[n0]; b0.y = vr1[n0];
            v2f b1; b1.x = vr0[n1]; b1.y = vr1[n1];
            acc0 = __builtin_amdgcn_wmma_f32_16x16x4_f32(false, a, false, b0,
                                                         (short)0, acc0, false, false);
            acc1 = __builtin_amdgcn_wmma_f32_16x16x4_f32(false, a, false, b1,
                                                         (short)0, acc1, false, false);
        }
        #pragma unroll
        for (int r = 0; r < 8; ++r) {
            const int m = r + 8 * kh;
            Os[((wave * 2 + 0) * TQ + m) * 16 + l16] = acc0[r];
            Os[((wave * 2 + 1) * TQ + m) * 16 + l16] = acc1[r];
        }
    }
    __syncthreads();

    // ---- combine the 4 K-quarters, write O (coalesced) ----
    for (int i = tid; i < TQ * DD; i += 256) {
        const int m = i / DD, n = i % DD;
        const int nt = n >> 4, nc = n & 15;
        const int tg = nt >> 1, tl = nt & 1;          // which tile group / local tile
        float val = 0.0f;
        #pragma unroll
        for (int kq = 0; kq < 4; ++kq) {
            const int w = kq * 2 + tg;
            val += Os[((w * 2 + tl) * TQ + m) * 16 + nc];
        }
        obase[m * DD + n] = val;
    }
}

extern "C" void kernel_launch(void* const* d_in, const int* in_sizes, int n_in,
                              void* d_out, int out_size, void* d_ws, size_t ws_size,
                              hipStream_t stream) {
    const float* q    = (const float*)d_in[0];
    const float* k    = (const float*)d_in[1];
    const float* v    = (const float*)d_in[2];
    const int*   mask = (const int*)d_in[3];

    float* out   = (float*)d_out;                              // [B,H,L,D]
    float* score = (float*)d_out + (size_t)BB * HH * LL * DD;  // [B,H,L,L]

    const int grid = BB * HH * (LL / TQ);  // 4096 workgroups
    attn_f32_wmma_kernel<<<grid, 256, 0, stream>>>(q, k, v, mask, out, score);
}